// TreeGNN_68487548502156
// MI455X (gfx1250) — compile-verified
//
#include <hip/hip_runtime.h>
#include <hip/hip_bf16.h>

typedef __attribute__((ext_vector_type(2))) float v2f;
typedef __attribute__((ext_vector_type(8))) float v8f;

#define HDIM 64
#define FOUT 16

// ---------------- init: deg=1 (self loop), zero pooling buffers ----------------
__global__ void gnn_init_kernel(float* deg, float* gsum, float* gcnt, int N, int G) {
    int i = blockIdx.x * blockDim.x + threadIdx.x;
    if (i < N) deg[i] = 1.0f;          // self-loop contributes 1 to every degree
    if (i < G * HDIM) gsum[i] = 0.0f;
    if (i < G) gcnt[i] = 0.0f;
}

// ---------------- degree accumulation over real edges ----------------
__global__ void gnn_deg_kernel(const int* dst, float* deg, int E) {
    int e = blockIdx.x * blockDim.x + threadIdx.x;
    if (e < E) atomicAdd(&deg[dst[e]], 1.0f);
}

// ---------------- dinv = rsqrt(deg); s = self-loop term of scalar prop ----------------
__global__ void gnn_dinv_kernel(const float* x, const float* deg, float* dinv, float* s, int N) {
    int n = blockIdx.x * blockDim.x + threadIdx.x;
    if (n < N) {
        float di = rsqrtf(deg[n]);
        dinv[n] = di;
        s[n] = x[n] * di * di;          // norm(self) = dinv^2
    }
}

// ---------------- layer-1 propagation: scalar per edge (Fin=1 commute trick) ----------------
__global__ void gnn_edge_scalar_kernel(const int* src, const int* dst, const float* x,
                                       const float* dinv, float* s, int E) {
    int e = blockIdx.x * blockDim.x + threadIdx.x;
    if (e < E) {
        int a = src[e], b = dst[e];
        atomicAdd(&s[b], dinv[a] * dinv[b] * x[a]);
    }
}

// ---------------- h1 = relu(s*W1 + b1); agg = dinv^2 * h1 (layer-2 self-loop seed) ----------------
__global__ void gnn_layer1_kernel(const float* s, const float* W1, const float* b1,
                                  const float* dinv, float* h1, float* agg, int N) {
    int t = blockIdx.x * blockDim.x + threadIdx.x;
    int n = t >> 6, j = t & 63;
    if (n >= N) return;
    float v = fmaxf(s[n] * W1[j] + b1[j], 0.0f);
    h1[t] = v;
    float di = dinv[n];
    agg[t] = di * di * v;
}

// ---------------- layer-2 edge propagation: 16 lanes/edge, float4 gather + f32 atomics ----------------
__global__ void gnn_edge_prop_kernel(const int* src, const int* dst, const float* dinv,
                                     const float* h1, float* agg, int E) {
    int t = blockIdx.x * blockDim.x + threadIdx.x;
    int e = t >> 4;
    if (e >= E) return;
    int q = (t & 15) << 2;                       // feature offset 0..60
    int a = src[e], b = dst[e];
    float w = dinv[a] * dinv[b];
    const float4 hv = *reinterpret_cast<const float4*>(h1 + (size_t)a * HDIM + q);
    float* o = agg + (size_t)b * HDIM + q;
    atomicAdd(o + 0, w * hv.x);
    atomicAdd(o + 1, w * hv.y);
    atomicAdd(o + 2, w * hv.z);
    atomicAdd(o + 3, w * hv.w);
}

// ---------------- WMMA fp32 GEMM: C[N x 64] = A[N x 64] @ B[64 x 64] + bias ----------------
// One wave32 per 16x16 output tile; V_WMMA_F32_16X16X4_F32, K unrolled by 4 (16 WMMAs).
// Row index is CLAMPED (not predicated) so the inner loop has no EXEC manipulation:
// out-of-range lanes recompute row N-1 and are simply not stored.
__global__ __launch_bounds__(32) void gnn_gemm64_wmma_kernel(const float* __restrict__ A,
                                                             const float* __restrict__ B,
                                                             const float* __restrict__ bias,
                                                             float* __restrict__ C, int N) {
    const int lane  = threadIdx.x;               // 0..31
    const int m0    = blockIdx.x * 16;
    const int n0    = blockIdx.y * 16;
    const int mrow  = min(m0 + (lane & 15), N - 1);
    const int ncol  = n0 + (lane & 15);
    const int khalf = (lane < 16) ? 0 : 2;       // A/B K pairs split across half-waves
    const float* arow = A + (size_t)mrow * HDIM + khalf;
    v8f acc = {};
#pragma unroll
    for (int k = 0; k < HDIM; k += 4) {
        v2f a = *reinterpret_cast<const v2f*>(arow + k);          // global_load_b64
        v2f b;
        const float* bp = B + (size_t)(k + khalf) * HDIM + ncol;
        b.x = bp[0];
        b.y = bp[HDIM];
        acc = __builtin_amdgcn_wmma_f32_16x16x4_f32(false, a, false, b, (short)0, acc,
                                                    false, false);
    }
    const float bb = bias[ncol];
    float* crow = C + (size_t)(m0 + ((lane < 16) ? 0 : 8)) * HDIM + ncol;
    if (m0 + 16 <= N) {                           // full tile: unconditional stores
#pragma unroll
        for (int r = 0; r < 8; ++r) crow[(size_t)r * HDIM] = acc[r] + bb;
    } else {                                      // tail tile: guarded stores
#pragma unroll
        for (int r = 0; r < 8; ++r) {
            int m = m0 + r + ((lane < 16) ? 0 : 8);
            if (m < N) C[(size_t)m * HDIM + ncol] = acc[r] + bb;
        }
    }
}

// ---------------- mean-pool accumulation ----------------
__global__ void gnn_pool_kernel(const float* h2, const int* batch, float* gsum, float* gcnt, int N) {
    int t = blockIdx.x * blockDim.x + threadIdx.x;
    int n = t >> 6, j = t & 63;
    if (n >= N) return;
    int g = batch[n];
    atomicAdd(&gsum[(size_t)g * HDIM + j], h2[t]);
    if (j == 0) atomicAdd(&gcnt[g], 1.0f);
}

// ---------------- final WMMA GEMM: out[G x 16] = (gsum/cnt)[G x 64] @ Wf[64 x 16] + bf ----------------
__global__ __launch_bounds__(32) void gnn_head_wmma_kernel(const float* __restrict__ gsum,
                                                           const float* __restrict__ gcnt,
                                                           const float* __restrict__ Wf,
                                                           const float* __restrict__ bf,
                                                           float* __restrict__ out, int G) {
    const int lane  = threadIdx.x;
    const int m0    = blockIdx.x * 16;
    const int grow  = min(m0 + (lane & 15), G - 1);
    const int ncol  = lane & 15;
    const int khalf = (lane < 16) ? 0 : 2;
    const float invc = 1.0f / fmaxf(gcnt[grow], 1.0f);
    const float* arow = gsum + (size_t)grow * HDIM + khalf;
    v8f acc = {};
#pragma unroll
    for (int k = 0; k < HDIM; k += 4) {
        v2f a = *reinterpret_cast<const v2f*>(arow + k);          // global_load_b64
        a.x *= invc;
        a.y *= invc;
        v2f b;
        const float* bp = Wf + (size_t)(k + khalf) * FOUT + ncol;
        b.x = bp[0];
        b.y = bp[FOUT];
        acc = __builtin_amdgcn_wmma_f32_16x16x4_f32(false, a, false, b, (short)0, acc,
                                                    false, false);
    }
    const float bb = bf[ncol];
    float* orow = out + (size_t)(m0 + ((lane < 16) ? 0 : 8)) * FOUT + ncol;
    if (m0 + 16 <= G) {
#pragma unroll
        for (int r = 0; r < 8; ++r) orow[(size_t)r * FOUT] = acc[r] + bb;
    } else {
#pragma unroll
        for (int r = 0; r < 8; ++r) {
            int g = m0 + r + ((lane < 16) ? 0 : 8);
            if (g < G) out[(size_t)g * FOUT + ncol] = acc[r] + bb;
        }
    }
}

extern "C" void kernel_launch(void* const* d_in, const int* in_sizes, int n_in,
                              void* d_out, int out_size, void* d_ws, size_t ws_size,
                              hipStream_t stream) {
    const float* x     = (const float*)d_in[0];   // [N,1]
    const int*   edge  = (const int*)d_in[1];     // [2,E] row-major
    const int*   batch = (const int*)d_in[2];     // [N]
    // d_in[3] = num_graphs (device scalar) — derived from out_size instead
    const float* W1 = (const float*)d_in[4];
    const float* b1 = (const float*)d_in[5];
    const float* W2 = (const float*)d_in[6];
    const float* b2 = (const float*)d_in[7];
    const float* Wf = (const float*)d_in[8];
    const float* bf = (const float*)d_in[9];

    const int N = in_sizes[0];
    const int E = in_sizes[1] / 2;
    const int G = out_size / FOUT;

    const int* src = edge;
    const int* dst = edge + E;

    // workspace carve-up (floats)
    float* ws   = (float*)d_ws;
    float* deg  = ws;                          // N
    float* dinv = deg + N;                     // N
    float* s    = dinv + N;                    // N
    float* h1   = s + N;                       // N*H  (reused as h2 after GEMM)
    float* agg  = h1 + (size_t)N * HDIM;       // N*H
    float* gsum = agg + (size_t)N * HDIM;      // G*H
    float* gcnt = gsum + (size_t)G * HDIM;     // G

    const int B = 256;
    int initM = N > G * HDIM ? N : G * HDIM;

    gnn_init_kernel<<<(initM + B - 1) / B, B, 0, stream>>>(deg, gsum, gcnt, N, G);
    gnn_deg_kernel<<<(E + B - 1) / B, B, 0, stream>>>(dst, deg, E);
    gnn_dinv_kernel<<<(N + B - 1) / B, B, 0, stream>>>(x, deg, dinv, s, N);
    gnn_edge_scalar_kernel<<<(E + B - 1) / B, B, 0, stream>>>(src, dst, x, dinv, s, E);

    size_t NH = (size_t)N * HDIM;
    gnn_layer1_kernel<<<(unsigned)((NH + B - 1) / B), B, 0, stream>>>(s, W1, b1, dinv, h1, agg, N);

    size_t EP = (size_t)E * 16;
    gnn_edge_prop_kernel<<<(unsigned)((EP + B - 1) / B), B, 0, stream>>>(src, dst, dinv, h1, agg, E);

    // h2 = agg @ W2 + b2   (WMMA fp32), write into h1 buffer (no longer needed)
    dim3 ggrid((N + 15) / 16, HDIM / 16);
    gnn_gemm64_wmma_kernel<<<ggrid, 32, 0, stream>>>(agg, W2, b2, h1, N);

    gnn_pool_kernel<<<(unsigned)((NH + B - 1) / B), B, 0, stream>>>(h1, batch, gsum, gcnt, N);

    gnn_head_wmma_kernel<<<(G + 15) / 16, 32, 0, stream>>>(gsum, gcnt, Wf, bf, (float*)d_out, G);
}